// YoloLoss_81595788689575
// MI455X (gfx1250) — compile-verified
//
#include <hip/hip_runtime.h>
#include <math.h>

// ---------------- problem constants (from reference) ----------------
#define Bc 32
#define Ac 5
#define Cc 80
#define Wc 52
#define Hc 52
#define Gc (Wc * Hc)        // 2704
#define Fc (5 + Cc)         // 85
#define Mc 20
#define BLOCKSZ 16.0f
#define THRESH 0.6f
#define OBJECT_SCALE 5.0f

#define AG (Ac * Gc)        // 13520 per-batch cells (NOT a multiple of 32)
#define KCONF_BLK 256
#define KCONF_GRIDX ((AG + KCONF_BLK - 1) / KCONF_BLK)   // 53

// ---------------- workspace layout (floats from base of d_ws) -------
#define WS_PART  0
#define WS_GTC   16
#define WS_TVAL  (WS_GTC + Bc * Mc * 4)      // 2576
#define WS_TCONF (WS_TVAL + Bc * Mc * 4)     // 5136
#define WS_KEY   (WS_TCONF + Bc * Mc)        // 5776
#define WS_TCLS  (WS_KEY + Bc * Mc)          // 6416

typedef __attribute__((ext_vector_type(2))) float v2f;
typedef __attribute__((ext_vector_type(8))) float v8f;

__device__ __forceinline__ float sigf(float x) { return 1.0f / (1.0f + __expf(-x)); }

__device__ __forceinline__ float rdlane(float v, int lane) {
    return __uint_as_float(__builtin_amdgcn_readlane(__float_as_uint(v), lane));
}

__device__ __forceinline__ float iou_box(float ax, float ay, float aw, float ah,
                                         float bx, float by, float bw, float bh) {
    float l = fmaxf(ax - aw * 0.5f, bx - bw * 0.5f);
    float r = fminf(ax + aw * 0.5f, bx + bw * 0.5f);
    float t = fmaxf(ay - ah * 0.5f, by - bh * 0.5f);
    float d = fminf(ay + ah * 0.5f, by + bh * 0.5f);
    float iw = fmaxf(r - l, 0.0f);
    float ih = fmaxf(d - t, 0.0f);
    float inter = iw * ih;
    return inter / (aw * ah + bw * bh - inter);
}

// ---------------- kernel 0: zero the partial accumulators ----------------
__global__ void k_init(float* ws) {
    if (threadIdx.x < 3) ws[WS_PART + threadIdx.x] = 0.0f;
}

// ---------------- kernel 1: per-GT prep (B blocks x 32 threads) ----------
__global__ void k_gtprep(const float* __restrict__ cur,
                         const float* __restrict__ gt,
                         const float* __restrict__ anchors,
                         float* __restrict__ ws) {
    int b = blockIdx.x;
    int m = threadIdx.x;
    if (m >= Mc) return;

    const float* g5 = gt + (b * Mc + m) * 5;
    float x = g5[0], y = g5[1], w = g5[2], h = g5[3];
    float gx = (x + w * 0.5f) / BLOCKSZ;
    float gy = (y + h * 0.5f) / BLOCKSZ;
    float gw = w / BLOCKSZ;
    float gh = h / BLOCKSZ;

    float* gtc = ws + WS_GTC + (b * Mc + m) * 4;
    gtc[0] = gx; gtc[1] = gy; gtc[2] = gw; gtc[3] = gh;

    // best anchor by wh-IOU (centers at 0); first index wins ties
    int best = 0;
    float bestIoU = -1.0f;
    for (int a = 0; a < Ac; ++a) {
        float aw = anchors[2 * a], ah = anchors[2 * a + 1];
        float inter = fminf(gw, aw) * fminf(gh, ah);
        float u = gw * gh + aw * ah - inter;
        float iou = inter / u;
        if (iou > bestIoU) { bestIoU = iou; best = a; }
    }

    int gi = (int)gx; gi = gi < 0 ? 0 : (gi > Wc - 1 ? Wc - 1 : gi);
    int gj = (int)gy; gj = gj < 0 ? 0 : (gj > Hc - 1 ? Hc - 1 : gj);
    int cell = gj * Wc + gi;

    float aw = anchors[2 * best], ah = anchors[2 * best + 1];
    float* tv = ws + WS_TVAL + (b * Mc + m) * 4;
    tv[0] = gx - (float)gi;
    tv[1] = gy - (float)gj;
    tv[2] = __logf(fmaxf(gw, 1.0f) / aw);
    tv[3] = __logf(fmaxf(gh, 1.0f) / ah);

    // tconf = IOU(gt cell box, stop-grad pred box at (best, cell))
    long base = ((long)(b * Ac + best) * Fc) * Gc + cell;
    float r0 = cur[base + 0L * Gc];
    float r1 = cur[base + 1L * Gc];
    float r2 = cur[base + 2L * Gc];
    float r3 = cur[base + 3L * Gc];
    float px = sigf(r0) + (float)(cell % Wc);
    float py = sigf(r1) + (float)(cell / Wc);
    float pw = __expf(__expf(r2)) * aw;   // reference double-exp
    float ph = __expf(__expf(r3)) * ah;
    ws[WS_TCONF + b * Mc + m] = iou_box(gx, gy, gw, gh, px, py, pw, ph);

    ((int*)ws)[WS_KEY + b * Mc + m] = best * Gc + cell;
    ((int*)ws)[WS_TCLS + b * Mc + m] = (int)g5[4];
}

// ---------------- kernel 2: conf loss; WMMA wave reduce --------------------
// grid = (53, B) x 256. b = blockIdx.y => all GT loads are wave-uniform
// (scalar-load friendly). Tail lanes stay ACTIVE with acc=0 and clamped
// addresses so EXEC is all-ones at the WMMA (ISA requirement).
__global__ void __launch_bounds__(KCONF_BLK) k_conf(
        const float* __restrict__ cur,
        const float* __restrict__ anchors,
        const float* __restrict__ gtc_all,    // ws + WS_GTC
        const int*   __restrict__ keys_all,   // ws + WS_KEY
        const float* __restrict__ tcf_all,    // ws + WS_TCONF
        float* __restrict__ part) {           // ws + WS_PART
    int b = blockIdx.y;
    int idx = blockIdx.x * KCONF_BLK + threadIdx.x;   // cell index within batch
    bool valid = idx < AG;
    int cidx = valid ? idx : (AG - 1);                // clamp for safe loads
    int a = cidx / Gc;
    int g = cidx % Gc;

    long base = ((long)(b * Ac + a) * Fc) * Gc + g;
    float r0 = cur[base + 0L * Gc];
    float r1 = cur[base + 1L * Gc];
    float r2 = cur[base + 2L * Gc];
    float r3 = cur[base + 3L * Gc];
    float r4 = cur[base + 4L * Gc];

    float aw = anchors[2 * a], ah = anchors[2 * a + 1];
    float px = sigf(r0) + (float)(g % Wc);
    float py = sigf(r1) + (float)(g / Wc);
    float pw = __expf(__expf(r2)) * aw;
    float ph = __expf(__expf(r3)) * ah;

    // noobj: any GT with IOU > THRESH (uniform GT addresses, no breaks)
    bool noobj = false;
    const float* gtc = gtc_all + b * (Mc * 4);
    #pragma unroll 4
    for (int m = 0; m < Mc; ++m) {
        float iou = iou_box(gtc[4 * m + 0], gtc[4 * m + 1],
                            gtc[4 * m + 2], gtc[4 * m + 3],
                            px, py, pw, ph);
        noobj = noobj || (iou > THRESH);
    }

    // special-cell override (forward scan => last writer wins for tconf)
    int key = a * Gc + g;
    const int*   keys = keys_all + b * Mc;
    const float* tcf  = tcf_all  + b * Mc;
    bool spec = false;
    float tconf = 0.0f;
    for (int m = 0; m < Mc; ++m) {
        bool hit = (keys[m] == key);
        spec = spec || hit;
        tconf = hit ? tcf[m] : tconf;
    }

    float mask = spec ? OBJECT_SCALE : (noobj ? 0.0f : 1.0f);
    float sm = __fsqrt_rn(mask);
    float conf = sigf(r4);
    float d = conf * sm - tconf * sm;
    float acc = valid ? d * d : 0.0f;

    // ---- wave32 reduction via V_WMMA_F32_16X16X4_F32 against ones ----
    // A (16x4): lanes 0-15 hold K=0/1, lanes 16-31 hold K=2/3.
    // B = ones(4x16) => D[m][*] = acc[lane m] + acc[lane m+16].
    v2f av; av.x = acc;  av.y = 0.0f;
    v2f bv; bv.x = 1.0f; bv.y = 1.0f;
    v8f cv = {};
    v8f dv = __builtin_amdgcn_wmma_f32_16x16x4_f32(false, av, false, bv,
                                                   (short)0, cv, false, false);
    // D vgpr r: lanes 0-15 = row r, lanes 16-31 = row r+8
    float s = dv[0] + dv[1] + dv[2] + dv[3] + dv[4] + dv[5] + dv[6] + dv[7];
    float tot = rdlane(s, 0) + rdlane(s, 16);   // rows 0-7 + rows 8-15
    if ((threadIdx.x & 31) == 0) atomicAdd(&part[1], tot);
}

// ---------------- kernel 3: coord + class loss at masked cells -------------
// grid = B*M blocks x 128 threads
__global__ void __launch_bounds__(128) k_masked(const float* __restrict__ cur,
                                                float* __restrict__ ws) {
    int b = blockIdx.x / Mc;
    int m = blockIdx.x % Mc;
    const int* keys = (const int*)ws + WS_KEY + b * Mc;
    int key = keys[m];

    // scatter-set duplicate resolution: only the last writer contributes
    bool last = true;
    for (int m2 = m + 1; m2 < Mc; ++m2) last = last && (keys[m2] != key);
    if (!last) return;   // uniform across block

    int a = key / Gc;
    int g = key % Gc;
    long base = ((long)(b * Ac + a) * Fc) * Gc + g;

    __shared__ float red[128];
    int t = threadIdx.x;

    // class logits (C=80 < 128 threads)
    float logit = -INFINITY;
    if (t < Cc) logit = cur[base + (long)(5 + t) * Gc];

    red[t] = logit;
    __syncthreads();
    for (int s2 = 64; s2 > 0; s2 >>= 1) {
        if (t < s2) red[t] = fmaxf(red[t], red[t + s2]);
        __syncthreads();
    }
    float mx = red[0];
    __syncthreads();

    red[t] = (t < Cc) ? __expf(logit - mx) : 0.0f;
    __syncthreads();
    for (int s2 = 64; s2 > 0; s2 >>= 1) {
        if (t < s2) red[t] += red[t + s2];
        __syncthreads();
    }

    if (t == 0) {
        float lse = __logf(red[0]) + mx;
        int tc = ((const int*)ws)[WS_TCLS + b * Mc + m];
        float lt = cur[base + (long)(5 + tc) * Gc];
        float nll = lse - lt;                       // -log_softmax[tc]
        atomicAdd(&ws[WS_PART + 2], 2.0f * nll);    // CLASS_SCALE * 2.0

        const float* tv = ws + WS_TVAL + (b * Mc + m) * 4;
        float cs = 0.0f;
        #pragma unroll
        for (int ch = 0; ch < 4; ++ch) {
            float raw = cur[base + (long)ch * Gc];
            float c = (ch < 2) ? sigf(raw) : __expf(raw);
            float df = c - tv[ch];                  // coord_mask == 1 here
            cs += df * df;
        }
        atomicAdd(&ws[WS_PART + 0], cs);            // COORD_SCALE == 1
    }
}

// ---------------- kernel 4: finalize -------------------------------------
__global__ void k_final(const float* __restrict__ ws, float* __restrict__ out) {
    if (threadIdx.x == 0 && blockIdx.x == 0) {
        float inv = 1.0f / (float)Bc;
        float lc = ws[WS_PART + 0] * inv;
        float lf = ws[WS_PART + 1] * inv;
        float ls = ws[WS_PART + 2] * inv;
        out[0] = lc + lf + ls;
        out[1] = lc;
        out[2] = lf;
        out[3] = ls;
    }
}

extern "C" void kernel_launch(void* const* d_in, const int* in_sizes, int n_in,
                              void* d_out, int out_size, void* d_ws, size_t ws_size,
                              hipStream_t stream) {
    const float* cur     = (const float*)d_in[0];   // (B, A, F, G)
    const float* gt      = (const float*)d_in[1];   // (B, M, 5)
    const float* anchors = (const float*)d_in[2];   // (A, 2)
    float* ws  = (float*)d_ws;
    float* out = (float*)d_out;

    k_init  <<<1, 32, 0, stream>>>(ws);
    k_gtprep<<<Bc, 32, 0, stream>>>(cur, gt, anchors, ws);

    dim3 cgrid(KCONF_GRIDX, Bc);
    k_conf  <<<cgrid, KCONF_BLK, 0, stream>>>(cur, anchors,
                                              ws + WS_GTC,
                                              (const int*)ws + WS_KEY,
                                              ws + WS_TCONF,
                                              ws + WS_PART);

    k_masked<<<Bc * Mc, 128, 0, stream>>>(cur, ws);
    k_final <<<1, 32, 0, stream>>>(ws, out);
}